// GCN_12867722019091
// MI455X (gfx1250) — compile-verified
//
#include <hip/hip_runtime.h>

typedef __attribute__((ext_vector_type(2))) float v2f;
typedef __attribute__((ext_vector_type(8))) float v8f;

// ---- K0: init deg (=1 for self loop) and u (=0) -------------------------------
__global__ void gcn_init(float* __restrict__ deg, float* __restrict__ u, int n) {
  int i = blockIdx.x * blockDim.x + threadIdx.x;
  if (i < n) { deg[i] = 1.0f; u[i] = 0.0f; }
}

// ---- K1: in-degree count over edge targets -----------------------------------
__global__ void gcn_degree(const int* __restrict__ col, float* __restrict__ deg, int e) {
  int i = blockIdx.x * blockDim.x + threadIdx.x;
  if (i < e) atomicAdd(&deg[col[i]], 1.0f);
}

// ---- K2: dinv = rsqrt(deg) in place (deg >= 1 always, self-loop) -------------
__global__ void gcn_rsqrt(float* __restrict__ deg, int n) {
  int i = blockIdx.x * blockDim.x + threadIdx.x;
  if (i < n) deg[i] = rsqrtf(deg[i]);
}

// ---- K3: u[col] += x[row] * dinv[row]  (scalar layer-1 aggregation) ----------
__global__ void gcn_agg1(const int* __restrict__ row, const int* __restrict__ col,
                         const float* __restrict__ x, const float* __restrict__ dinv,
                         float* __restrict__ u, int e) {
  int i = blockIdx.x * blockDim.x + threadIdx.x;
  if (i < e) {
    int r = row[i];
    atomicAdd(&u[col[i]], x[r] * dinv[r]);
  }
}

// ---- K4: per-node dense transform via WMMA f32 16x16x4 -----------------------
// s[i]   = dinv[i]*u[i] + dinv[i]^2 * x[i]
// A[m,k] = relu(s[m]*W1[k] + b1[k])          (16x64, built in-register)
// B[k,n] = n<2 ? W2[k,n] : 0                 (64x16 zero-padded, branch-free)
// D      = A @ B  -> h2[node, 0:2]
// out[node,n] = dinv^2 * h2 + b2[n]          (self-loop + bias init)
__global__ void gcn_dense_wmma(const float* __restrict__ x, const float* __restrict__ dinv,
                               const float* __restrict__ u,
                               const float* __restrict__ W1, const float* __restrict__ b1,
                               const float* __restrict__ W2, const float* __restrict__ b2,
                               float* __restrict__ h2, float* __restrict__ out,
                               int n, int ntiles) {
  int wave = threadIdx.x >> 5;
  int tile = blockIdx.x * (blockDim.x >> 5) + wave;
  if (tile >= ntiles) return;            // uniform per wave -> EXEC all-ones for WMMA
  int lane  = threadIdx.x & 31;
  int m     = lane & 15;                 // row (A half) / column (B, D)
  int khalf = (lane >> 4) * 2;           // K sub-slot: lanes 0-15 -> {0,1}, 16-31 -> {2,3}

  int node = tile * 16 + m;
  int nc   = node < n ? node : n - 1;    // clamp for partial last tile (rows unused)
  float dv = dinv[nc];
  float sv = dv * u[nc] + dv * dv * x[nc];

  // Branch-free W2 access: clamp column index, zero pad columns via mask multiply.
  float msk = (m < 2) ? 1.0f : 0.0f;     // v_cndmask, no exec divergence
  int   ci  = (m < 2) ? m : 0;

  v8f acc = {0.f, 0.f, 0.f, 0.f, 0.f, 0.f, 0.f, 0.f};
  #pragma unroll
  for (int k0 = 0; k0 < 64; k0 += 4) {
    int k = k0 + khalf;
    v2f a, b;
    a.x = fmaxf(fmaf(sv, W1[k],     b1[k]),     0.f);
    a.y = fmaxf(fmaf(sv, W1[k + 1], b1[k + 1]), 0.f);
    b.x = msk * W2[k * 2 + ci];            // unconditional loads, masked in VALU
    b.y = msk * W2[(k + 1) * 2 + ci];
    acc = __builtin_amdgcn_wmma_f32_16x16x4_f32(
        /*neg_a=*/false, a, /*neg_b=*/false, b,
        /*c_mod=*/(short)0, acc, /*reuse_a=*/false, /*reuse_b=*/false);
  }

  // D layout: VGPR j holds rows M=j (lanes 0-15) / M=j+8 (lanes 16-31), N = lane&15
  if (m < 2) {
    int half = lane >> 4;
    float bb = b2[m];
    #pragma unroll
    for (int j = 0; j < 8; ++j) {
      int mm = j + half * 8;
      int nd = tile * 16 + mm;
      if (nd < n) {
        float v = acc[j];
        h2[nd * 2 + m]  = v;
        float dn = dinv[nd];
        out[nd * 2 + m] = dn * dn * v + bb;   // self-loop term + bias
      }
    }
  }
}

// ---- K5: out[col,:] += dinv[row]*dinv[col] * h2[row,:] -----------------------
__global__ void gcn_agg2(const int* __restrict__ row, const int* __restrict__ col,
                         const float* __restrict__ dinv, const float* __restrict__ h2,
                         float* __restrict__ out, int e) {
  int i = blockIdx.x * blockDim.x + threadIdx.x;
  if (i < e) {
    int r = row[i], c = col[i];
    float w = dinv[r] * dinv[c];
    float2 hv = ((const float2*)h2)[r];
    atomicAdd(&out[c * 2 + 0], w * hv.x);
    atomicAdd(&out[c * 2 + 1], w * hv.y);
  }
}

extern "C" void kernel_launch(void* const* d_in, const int* in_sizes, int n_in,
                              void* d_out, int out_size, void* d_ws, size_t ws_size,
                              hipStream_t stream) {
  const float* x  = (const float*)d_in[0];
  const int*   ei = (const int*)d_in[1];
  const float* W1 = (const float*)d_in[2];
  const float* b1 = (const float*)d_in[3];
  const float* W2 = (const float*)d_in[4];
  const float* b2 = (const float*)d_in[5];
  float* out = (float*)d_out;

  int n = in_sizes[0];          // NUM_FEATURES == 1 -> flat size == N_NODES
  int e = in_sizes[1] / 2;      // edge_index is [2, E]
  const int* row = ei;
  const int* col = ei + e;

  float* dinv = (float*)d_ws;   // n floats (deg, then rsqrt in place)
  float* u    = dinv + n;       // n floats
  float* h2   = u + n;          // 2n floats

  const int B  = 256;
  int gn = (n + B - 1) / B;
  int ge = (e + B - 1) / B;

  gcn_init  <<<gn, B, 0, stream>>>(dinv, u, n);
  gcn_degree<<<ge, B, 0, stream>>>(col, dinv, e);
  gcn_rsqrt <<<gn, B, 0, stream>>>(dinv, n);
  gcn_agg1  <<<ge, B, 0, stream>>>(row, col, x, dinv, u, e);

  int ntiles = (n + 15) / 16;
  const int wavesPerBlock = 8;                  // 256 threads = 8 wave32
  int gt = (ntiles + wavesPerBlock - 1) / wavesPerBlock;
  gcn_dense_wmma<<<gt, wavesPerBlock * 32, 0, stream>>>(x, dinv, u, W1, b1, W2, b2,
                                                        h2, out, n, ntiles);

  gcn_agg2  <<<ge, B, 0, stream>>>(row, col, dinv, h2, out, e);
}